// PhysicsLayer_56444460204337
// MI455X (gfx1250) — compile-verified
//
#include <hip/hip_runtime.h>
#include <stdint.h>

// ---------------------------------------------------------------------------
// PhysicsLayer feature kernel for MI455X (gfx1250).
//
// Memory-bound streaming workload (~167 MB @ 23.3 TB/s ~= 7 us floor; ~1 GFLOP
// of math). No matmul structure -> WMMA is not applicable; the CDNA5 wins are
// (a) 320 KB LDS/WGP: a full 8192-step row of staged channels lives on-chip,
// (b) TDM async tensor DMA (tensor_load_to_lds) de-interleaves the stride-6
//     channels straight into LDS, tracked with TENSORcnt (s_wait_tensorcnt),
// (c) chunked affine-scan parallelization of the four EMA recurrences,
// (d) O(1)/element sliding windows: incremental sums + van Herk prefix/suffix
//     max-min for all four window extrema.
// ---------------------------------------------------------------------------

#define T_LEN   8192
#define NTHR    512
#define CHUNK   16          // T_LEN / NTHR
#define W_WIN   20
#define NBLK    ((T_LEN + W_WIN - 1) / W_WIN)   // 410 van-Herk blocks
#define EPSF    1e-8f
#define NEG_INF (-3.402823466e38f)
#define POS_INF ( 3.402823466e38f)

typedef __attribute__((ext_vector_type(4))) unsigned int tdm_v4u;
typedef __attribute__((ext_vector_type(4))) int          tdm_v4i;
typedef __attribute__((ext_vector_type(8))) int          tdm_v8i;

#if defined(__HIP_DEVICE_COMPILE__) && __has_builtin(__builtin_amdgcn_tensor_load_to_lds)
#define HAVE_TDM 1
#else
#define HAVE_TDM 0
#endif

#if defined(__HIP_DEVICE_COMPILE__) && __has_builtin(__builtin_amdgcn_s_wait_tensorcnt)
#define WAIT_TENSORCNT0() __builtin_amdgcn_s_wait_tensorcnt(0)
#else
#define WAIT_TENSORCNT0() do {} while (0)
#endif

#if HAVE_TDM
// DMA `nrows` f32 elements, spaced `row_stride_elems` apart in global memory,
// into a contiguous LDS array. D# packing per CDNA5 ISA ch.8 (Tensor DMA
// Descriptor): 2D tensor, tile = 1 x nrows, data_size = 4 bytes.
__device__ __forceinline__ void tdm_load_strided_f32(const void* gsrc, void* ldst,
                                                     uint32_t nrows,
                                                     uint32_t row_stride_elems)
{
    uint64_t ga = (uint64_t)(uintptr_t)gsrc;
    uint32_t la = (uint32_t)(uintptr_t)ldst;   // low 32 bits of flat = LDS byte addr

    tdm_v4u g0;
    g0.x = 1u;                                    // count=1 (valid), is_restore=0, gather off
    g0.y = la;                                    // D#.lds_addr
    g0.z = (uint32_t)ga;                          // global_addr[31:0]
    g0.w = (uint32_t)((ga >> 32) & 0x01FFFFFFu)   // global_addr[56:32]
         | (2u << 30);                            // type = 2 ("image")

    const uint32_t td0 = 1u;                      // tensor_dim0 (elements)
    const uint32_t td1 = nrows;                   // tensor_dim1
    const uint32_t tl0 = 1u;                      // tile_dim0
    const uint32_t tl1 = nrows;                   // tile_dim1 (16-bit field; 8192 ok)
    const uint32_t tl2 = 0u;
    const uint64_t st0 = row_stride_elems;        // tensor_dim0_stride
    const uint64_t st1 = 0u;

    tdm_v8i g1;
    g1[0] = (int)(2u << 16);                                            // data_size=2 -> 4B
    g1[1] = (int)((td0 & 0xFFFFu) << 16);                               // abar=0 | td0.lo16
    g1[2] = (int)(((td0 >> 16) & 0xFFFFu) | ((td1 & 0xFFFFu) << 16));
    g1[3] = (int)(((td1 >> 16) & 0xFFFFu) | ((tl0 & 0xFFFFu) << 16));
    g1[4] = (int)((tl1 & 0xFFFFu) | ((tl2 & 0xFFFFu) << 16));
    g1[5] = (int)(uint32_t)(st0 & 0xFFFFFFFFu);
    g1[6] = (int)(((uint32_t)((st0 >> 32) & 0xFFFFu)) |
                  (((uint32_t)(st1 & 0xFFFFu)) << 16));
    g1[7] = (int)(uint32_t)((st1 >> 16) & 0xFFFFFFFFu);

    tdm_v4i g2 = {0, 0, 0, 0};
    tdm_v4i g3 = {0, 0, 0, 0};
#if defined(__clang_major__) && (__clang_major__ >= 23)
    tdm_v8i g4 = {0, 0, 0, 0, 0, 0, 0, 0};
    __builtin_amdgcn_tensor_load_to_lds(g0, g1, g2, g3, g4, 0);
#else
    __builtin_amdgcn_tensor_load_to_lds(g0, g1, g2, g3, 0);
#endif
}
#endif  // HAVE_TDM

// van Herk building blocks: per aligned W_WIN block, prefix/suffix extremum.
__device__ __forceinline__ void block_ps_max(const float* __restrict__ src,
                                             float* __restrict__ P,
                                             float* __restrict__ S, int tid)
{
    for (int blk = tid; blk < NBLK; blk += NTHR) {
        const int st = blk * W_WIN;
        const int en = (st + W_WIN < T_LEN) ? (st + W_WIN) : T_LEN;
        float m = NEG_INF;
        for (int j = st; j < en; ++j) { m = fmaxf(m, src[j]); P[j] = m; }
        m = NEG_INF;
        for (int j = en - 1; j >= st; --j) { m = fmaxf(m, src[j]); S[j] = m; }
    }
}
__device__ __forceinline__ void block_ps_min(const float* __restrict__ src,
                                             float* __restrict__ P,
                                             float* __restrict__ S, int tid)
{
    for (int blk = tid; blk < NBLK; blk += NTHR) {
        const int st = blk * W_WIN;
        const int en = (st + W_WIN < T_LEN) ? (st + W_WIN) : T_LEN;
        float m = POS_INF;
        for (int j = st; j < en; ++j) { m = fminf(m, src[j]); P[j] = m; }
        m = POS_INF;
        for (int j = en - 1; j >= st; --j) { m = fminf(m, src[j]); S[j] = m; }
    }
}
// Exact for pad-left-repeat windows: t<W -> clamped window 0..t lies in block 0.
__device__ __forceinline__ float win_max_at(const float* P, const float* S, int t)
{
    return (t < W_WIN) ? P[t] : fmaxf(S[t - (W_WIN - 1)], P[t]);
}
__device__ __forceinline__ float win_min_at(const float* P, const float* S, int t)
{
    return (t < W_WIN) ? P[t] : fminf(S[t - (W_WIN - 1)], P[t]);
}

__global__ __launch_bounds__(NTHR)
void physics_features_kernel(const float* __restrict__ x, float* __restrict__ out)
{
    extern __shared__ float smem[];
    float* sHigh  = smem;               // [T_LEN] staged high; later: win-max of lc
    float* sLow   = smem + 1 * T_LEN;   // [T_LEN]
    float* sClose = smem + 2 * T_LEN;   // [T_LEN]
    float* sEma20 = smem + 3 * T_LEN;   // [T_LEN]
    float* sLc    = smem + 4 * T_LEN;   // [T_LEN] log(close+eps)
    float* sTr    = smem + 5 * T_LEN;   // [T_LEN] true range
    float* sPe    = smem + 6 * T_LEN;   // [T_LEN] temp win-max(high), then pe
    float* sP     = smem + 7 * T_LEN;   // [T_LEN] van-Herk prefix scratch
    float* sS     = smem + 8 * T_LEN;   // [T_LEN] van-Herk suffix scratch
    float* sScan  = smem + 9 * T_LEN;   // [4][NTHR][2] chain (decay,partial)/incoming

    const int tid = threadIdx.x;
    const int b   = blockIdx.x;
    const float* xrow = x + (size_t)b * T_LEN * 6;

    const float A21 = 2.0f / 21.0f;     // alpha = 2/(W+1)
    const float B21 = 1.0f - A21;

    // ---- Phase A: stage high/low/close/ema20 (stride-6 channels) into LDS --
#if HAVE_TDM
    if (tid < 32) {                     // wave 0 issues the DMAs (EXEC ignored by TDM)
        tdm_load_strided_f32(xrow + 1, sHigh,  T_LEN, 6);
        tdm_load_strided_f32(xrow + 2, sLow,   T_LEN, 6);
        tdm_load_strided_f32(xrow + 3, sClose, T_LEN, 6);
        tdm_load_strided_f32(xrow + 5, sEma20, T_LEN, 6);
        WAIT_TENSORCNT0();              // TENSORcnt==0: tiles landed in LDS
    }
#else
    for (int i = tid; i < T_LEN; i += NTHR) {
        const float* p = xrow + (size_t)i * 6;
        sHigh[i] = p[1]; sLow[i] = p[2]; sClose[i] = p[3]; sEma20[i] = p[5];
    }
#endif
    __syncthreads();

    // log-close and true-range, each computed exactly once
    for (int i = tid; i < T_LEN; i += NTHR) {
        sLc[i] = logf(sClose[i] + EPSF);
        const float hi = sHigh[i], lo = sLow[i];
        const float cp = sClose[(i - 1) < 0 ? 0 : i - 1];     // roll1(close)
        sTr[i] = fmaxf(fmaxf(hi - lo, fabsf(hi - cp)), fabsf(lo - cp));
    }
    __syncthreads();

    // ---- window max of high -> sPe (temp) ----------------------------------
    block_ps_max(sHigh, sP, sS, tid);
    __syncthreads();
    for (int i = tid; i < T_LEN; i += NTHR)
        sPe[i] = win_max_at(sP, sS, i);
    __syncthreads();
    // ---- window min of low: keep as P/S, consumed in Phase B ---------------
    block_ps_min(sLow, sP, sS, tid);
    __syncthreads();

    // ---- Phase B: per-chunk entropy + EMA chunk summaries ------------------
    {
        const int s = tid * CHUNK;
        float trsum = 0.f;      // rolling window sum of tr (pad == index clamp)
        for (int j = s - (W_WIN - 1); j <= s; ++j) trsum += sTr[j < 0 ? 0 : j];

        float cE = 0.f, cV = 0.f, cP = 0.f, cM = 0.f;   // chunk partials
        for (int i = 0; i < CHUNK; ++i) {
            const int t = s + i;
            if (i) trsum += sTr[t] - sTr[(t - W_WIN) < 0 ? 0 : (t - W_WIN)];
            const float hmax = sPe[t];                       // win_max(high)
            const float lmin = win_min_at(sP, sS, t);        // win_min(low)
            const float pe = log10f(trsum / (hmax - lmin + EPSF) + EPSF);
            sPe[t] = pe;                                     // overwrite in place (same thread)

            const float lct = sLc[t];
            const float lc1 = sLc[(t - 1) < 0 ? 0 : t - 1];
            const float lc2 = sLc[(t - 2) < 0 ? 0 : t - 2];
            const float ret = lct - lc1;                     // diff, prepend -> 0 at t=0
            const float vol = fabsf(ret);
            const float rp  = (t >= 1) ? (lc1 - lc2) : ret;  // roll1(ret)
            const float ld  = logf((vol + EPSF) / (fabsf(rp) + EPSF));

            if (t == 0) { cE = lct; cV = vol; cP = pe; cM = ld; }   // EMA init y0=x0
            else {
                cE = 0.1f * lct + 0.9f * cE;
                cV = A21 * vol + B21 * cV;
                cP = A21 * pe  + B21 * cP;
                cM = A21 * ld  + B21 * cM;
            }
        }
        float dE, dO;
        if (tid == 0) { dE = 0.f; dO = 0.f; }   // chunk 0: absolute (decay 0)
        else {
            dE = 1.f; dO = 1.f;
            for (int k = 0; k < CHUNK; ++k) { dE *= 0.9f; dO *= B21; }
        }
        sScan[0 * 2 * NTHR + tid * 2] = dE; sScan[0 * 2 * NTHR + tid * 2 + 1] = cE;
        sScan[1 * 2 * NTHR + tid * 2] = dO; sScan[1 * 2 * NTHR + tid * 2 + 1] = cV;
        sScan[2 * 2 * NTHR + tid * 2] = dO; sScan[2 * 2 * NTHR + tid * 2 + 1] = cP;
        sScan[3 * 2 * NTHR + tid * 2] = dO; sScan[3 * 2 * NTHR + tid * 2 + 1] = cM;
    }
    __syncthreads();

    // ---- Phase B2: serial affine combine (4 lanes) + lc prefix/suffix max --
    if (tid < 4) {
        const int base = tid * 2 * NTHR;
        float c = 0.f;
        for (int k = 0; k < NTHR; ++k) {
            const float D = sScan[base + 2 * k];
            const float P = sScan[base + 2 * k + 1];
            sScan[base + 2 * k] = c;      // incoming state for chunk k == y[s-1]
            c = D * c + P;
        }
    }
    block_ps_max(sLc, sP, sS, tid);       // disjoint arrays: overlaps with combine
    __syncthreads();
    for (int i = tid; i < T_LEN; i += NTHR)
        sHigh[i] = win_max_at(sP, sS, i); // sHigh recycled: win_max(lc)
    __syncthreads();
    block_ps_min(sLc, sP, sS, tid);       // win_min(lc) kept as P/S
    __syncthreads();

    // ---- Phase C: replay chunk with exact incoming states, emit features ---
    {
        const int s = tid * CHUNK;
        float cE = sScan[0 * 2 * NTHR + tid * 2];
        float cV = sScan[1 * 2 * NTHR + tid * 2];
        float cP = sScan[2 * 2 * NTHR + tid * 2];
        float cM = sScan[3 * 2 * NTHR + tid * 2];
        float res_prev = 0.f, pe_prev = 0.f, dent_prev = 0.f, mle_prev = 0.f, ediv_prev = 0.f;
        if (tid > 0) {
            res_prev  = sLc[s - 1] - cE;            // res[s-1] = lc - ekf_p
            pe_prev   = sPe[s - 1];
            dent_prev = sPe[s - 1] - sPe[s - 2];
            mle_prev  = cM;                          // mle value at s-1
            const float cl  = sClose[s - 1];
            const float e20 = sEma20[s - 1];
            ediv_prev = (cl - e20) / (e20 + EPSF);
        }

        // rolling Hurst sums over log-close
        float s1 = 0.f, s2 = 0.f;
        for (int j = s - (W_WIN - 1); j <= s; ++j) {
            const float v = sLc[j < 0 ? 0 : j];
            s1 += v; s2 += v * v;
        }

        const float HALF_LOG_W = 0.5f * logf(20.0f);
        for (int i = 0; i < CHUNK; ++i) {
            const int t = s + i;
            const float lct = sLc[t];
            const float lc1 = sLc[(t - 1) < 0 ? 0 : t - 1];
            const float lc2 = sLc[(t - 2) < 0 ? 0 : t - 2];
            const float ret = lct - lc1;
            const float vol = fabsf(ret);
            const float rp  = (t >= 1) ? (lc1 - lc2) : ret;
            const float ld  = logf((vol + EPSF) / (fabsf(rp) + EPSF));
            const float pe  = sPe[t];

            if (i) {
                const float vr = sLc[(t - W_WIN) < 0 ? 0 : (t - W_WIN)];
                s1 += lct - vr;
                s2 += lct * lct - vr * vr;
            }
            const float lmax = sHigh[t];                 // win_max(lc)
            const float lmn  = win_min_at(sP, sS, t);    // win_min(lc)
            const float mean = s1 * (1.0f / 20.0f);
            float var = (s2 - 20.0f * mean * mean) * (1.0f / 19.0f);
            var = fmaxf(var, 0.f);
            const float S = sqrtf(var) + EPSF;
            const float R = lmax - lmn + EPSF;
            const float dev = logf(R / S) - HALF_LOG_W;
            const float h = 0.5f + 0.3f * tanhf(dev * 2.0f);

            const float cl  = sClose[t];
            const float e20 = sEma20[t];
            const float ediv = (cl - e20) / (e20 + EPSF);

            float ekf_v, res, d_res, mle, d_mle, dent, dd, dediv, comp;
            if (t == 0) {
                cE = lct; cV = vol; cP = pe; cM = ld;
                ekf_v = 0.f; res = lct - cE; d_res = 0.f;
                mle = cM; d_mle = 0.f; dent = 0.f; dd = 0.f; dediv = 0.f;
                comp = fmaxf(cV - vol, 0.f) * fmaxf(cP - pe, 0.f);  // == 0
                res_prev = res; pe_prev = pe; dent_prev = dent;
                mle_prev = mle; ediv_prev = ediv;
            } else {
                const float eprev = cE;
                cE = 0.1f * lct + 0.9f * cE;
                ekf_v = cE - eprev;
                res = lct - cE; d_res = res - res_prev; res_prev = res;
                cM = A21 * ld + B21 * cM; mle = cM;
                d_mle = mle - mle_prev; mle_prev = mle;
                dent = pe - pe_prev; dd = dent - dent_prev;
                pe_prev = pe; dent_prev = dent;
                cV = A21 * vol + B21 * cV;
                cP = A21 * pe  + B21 * cP;
                comp = fmaxf(cV - vol, 0.f) * fmaxf(cP - pe, 0.f);
                dediv = ediv - ediv_prev; ediv_prev = ediv;
            }

            float* op = out + ((size_t)b * T_LEN + t) * 14;
            op[0]  = h;     op[1]  = vol;   op[2]  = ekf_v; op[3]  = res;
            op[4]  = pe;    op[5]  = mle;   op[6]  = ret;   op[7]  = ediv;
            op[8]  = dent;  op[9]  = dd;    op[10] = d_mle; op[11] = d_res;
            op[12] = dediv; op[13] = comp;
        }
    }
}

extern "C" void kernel_launch(void* const* d_in, const int* in_sizes, int n_in,
                              void* d_out, int out_size, void* d_ws, size_t ws_size,
                              hipStream_t stream)
{
    (void)n_in; (void)out_size; (void)d_ws; (void)ws_size;
    const float* x = (const float*)d_in[0];
    float* out = (float*)d_out;
    const int B = in_sizes[0] / (T_LEN * 6);                 // 256 rows
    const size_t shmem = (size_t)(9 * T_LEN + 4 * 2 * NTHR) * sizeof(float); // 311296 B

    // Opt-in to large dynamic LDS (gfx1250: up to 320 KB per workgroup).
    (void)hipFuncSetAttribute((const void*)physics_features_kernel,
                              hipFuncAttributeMaxDynamicSharedMemorySize,
                              (int)shmem);

    physics_features_kernel<<<dim3(B), dim3(NTHR), shmem, stream>>>(x, out);
}